// WGCN_10917806867120
// MI455X (gfx1250) — compile-verified
//
#include <hip/hip_runtime.h>
#include <math.h>

typedef float v2f __attribute__((ext_vector_type(2)));
typedef float v8f __attribute__((ext_vector_type(8)));

#define N_ENT_C 50000
#define DDIM 64

// ======================= edge scatter: msg[src] += att(rel)*x[dst] ==========
__global__ __launch_bounds__(256) void edge_msg_kernel(
    const int* __restrict__ trip, const float* __restrict__ re_att,
    const float* __restrict__ x, float* __restrict__ msg, int nE)
{
    const long total = (long)nE * 16;                 // 16 threads/edge, 4 ch each
    const long stride = (long)gridDim.x * blockDim.x;
    for (long t = (long)blockIdx.x * blockDim.x + threadIdx.x; t < total; t += stride) {
        const int e = (int)(t >> 4);
        const int q = ((int)t & 15) * 4;
        const int s = trip[3 * e + 0];
        const int r = trip[3 * e + 1];
        const int d = trip[3 * e + 2];
        const float att = re_att[r];
        const float4 xv = *(const float4*)(x + d * DDIM + q);
        float* mp = msg + s * DDIM + q;
        atomicAdd(mp + 0, xv.x * att);
        atomicAdd(mp + 1, xv.y * att);
        atomicAdd(mp + 2, xv.z * att);
        atomicAdd(mp + 3, xv.w * att);
    }
}

// ======================= GEMM: y = tanh((x+msg) @ W), W 64x64 row-major =====
// One wave computes a 16x64 output strip via V_WMMA_F32_16X16X4_F32 (K=4 steps).
// A 16x4 f32 fragment (ISA 7.12.2): lanes0-15: M=lane, v[0]=K0,v[1]=K1;
// lanes16-31: M=lane-16, v[0]=K2,v[1]=K3. B symmetric with N across lanes.
__global__ __launch_bounds__(256) void wgc_gemm_tanh(
    const float* __restrict__ xin, const float* __restrict__ msg,
    const float* __restrict__ W, float* __restrict__ yout, int nrows)
{
    __shared__ float sW[64 * 64];        // 16 KB
    __shared__ float sA[8 * 64 * 17];    // 34 KB: per-wave A^T tile, padded stride

    const int tid = threadIdx.x;
    const int wave = tid >> 5;
    const int lane = tid & 31;
    const int rowBase = blockIdx.x * 128 + wave * 16;

    { // cooperative load of W
        const float4* W4 = (const float4*)W;
        float4* sW4 = (float4*)sW;
        #pragma unroll
        for (int i = 0; i < 4; ++i) sW4[i * 256 + tid] = W4[i * 256 + tid];
    }

    // per-wave: fuse x+msg, store 16x64 tile transposed [k][m] (stride 17)
    float* sAT = &sA[wave * 64 * 17];
    #pragma unroll
    for (int i = 0; i < 8; ++i) {
        const int f = i * 32 + lane;          // float4 slot 0..255
        const int r = f >> 4;                 // tile row 0..15
        const int c4 = (f & 15) * 4;          // col 0,4,...,60
        int gr = rowBase + r; if (gr >= nrows) gr = nrows - 1;
        const float4 xv = *(const float4*)(xin + gr * DDIM + c4);
        const float4 mv = *(const float4*)(msg + gr * DDIM + c4);
        sAT[(c4 + 0) * 17 + r] = xv.x + mv.x;
        sAT[(c4 + 1) * 17 + r] = xv.y + mv.y;
        sAT[(c4 + 2) * 17 + r] = xv.z + mv.z;
        sAT[(c4 + 3) * 17 + r] = xv.w + mv.w;
    }
    __syncthreads();

    v8f acc0 = {}, acc1 = {}, acc2 = {}, acc3 = {};
    const int m  = lane & 15;
    const int kb = (lane >> 4) * 2;

    #pragma unroll
    for (int kk = 0; kk < 16; ++kk) {
        const int k = kk * 4 + kb;
        v2f a;  a.x  = sAT[k * 17 + m];        a.y  = sAT[(k + 1) * 17 + m];
        v2f b0; b0.x = sW[k * 64 +  0 + m];    b0.y = sW[(k + 1) * 64 +  0 + m];
        v2f b1; b1.x = sW[k * 64 + 16 + m];    b1.y = sW[(k + 1) * 64 + 16 + m];
        v2f b2; b2.x = sW[k * 64 + 32 + m];    b2.y = sW[(k + 1) * 64 + 32 + m];
        v2f b3; b3.x = sW[k * 64 + 48 + m];    b3.y = sW[(k + 1) * 64 + 48 + m];
        acc0 = __builtin_amdgcn_wmma_f32_16x16x4_f32(false, a, false, b0, (short)0, acc0, false, false);
        acc1 = __builtin_amdgcn_wmma_f32_16x16x4_f32(false, a, false, b1, (short)0, acc1, false, false);
        acc2 = __builtin_amdgcn_wmma_f32_16x16x4_f32(false, a, false, b2, (short)0, acc2, false, false);
        acc3 = __builtin_amdgcn_wmma_f32_16x16x4_f32(false, a, false, b3, (short)0, acc3, false, false);
    }

    const int rhalf = (lane >> 4) * 8;
    #pragma unroll
    for (int r = 0; r < 8; ++r) {
        const int row = rowBase + rhalf + r;
        if (row < nrows) {
            float* yr = yout + row * DDIM;
            yr[ 0 + m] = tanhf(acc0[r]);
            yr[16 + m] = tanhf(acc1[r]);
            yr[32 + m] = tanhf(acc2[r]);
            yr[48 + m] = tanhf(acc3[r]);
        }
    }
}

// ======= fold attention params: v_src=en_W^T u[:d], v_dst=en_W^T u[2d:], zmid
__global__ __launch_bounds__(64) void prep_kernel(
    const float* __restrict__ u, const float* __restrict__ enw,
    const float* __restrict__ rew, const float* __restrict__ respec,
    float* __restrict__ vsrc, float* __restrict__ vdst, float* __restrict__ sc)
{
    const int k = threadIdx.x;
    float a = 0.f, b = 0.f;
    for (int j = 0; j < 64; ++j) {
        const float w = enw[j * 64 + k];
        a += u[j] * w;
        b += u[128 + j] * w;
    }
    vsrc[k] = a; vdst[k] = b;

    __shared__ float red[64];
    float mdot = 0.f;
    for (int k2 = 0; k2 < 64; ++k2) mdot += rew[k * 64 + k2] * respec[k2];
    red[k] = u[64 + k] * mdot;
    __syncthreads();
    for (int s = 32; s > 0; s >>= 1) {
        if (k < s) red[k] += red[k + s];
        __syncthreads();
    }
    if (k == 0) sc[0] = red[0];   // z_mid
}

// ============ per-entity scores: z_src[i]=x[i].vsrc, z_dst[i]=x[i].vdst =====
__global__ __launch_bounds__(256) void z_kernel(
    const float* __restrict__ x, const float* __restrict__ vsrc,
    const float* __restrict__ vdst, float* __restrict__ zsrc,
    float* __restrict__ zdst, int n)
{
    const int i = blockIdx.x * blockDim.x + threadIdx.x;
    if (i >= n) return;
    float a = 0.f, b = 0.f;
    const float* xi = x + i * DDIM;
    #pragma unroll 4
    for (int k = 0; k < DDIM; ++k) {
        const float xv = xi[k];
        a += xv * vsrc[k];
        b += xv * vdst[k];
    }
    zsrc[i] = a; zdst[i] = b;
}

// ============ edge scores + per-block max ===================================
__global__ __launch_bounds__(256) void score_max_kernel(
    const int* __restrict__ trip, const float* __restrict__ zsrc,
    const float* __restrict__ zdst, const float* __restrict__ sc,
    float* __restrict__ scores, float* __restrict__ pmax, int nE)
{
    const float zmid = sc[0];
    float lm = -3.4e38f;
    const int stride = gridDim.x * blockDim.x;
    for (int e = blockIdx.x * blockDim.x + threadIdx.x; e < nE; e += stride) {
        const int s = trip[3 * e + 0];
        const int d = trip[3 * e + 2];
        float v = zsrc[s] + zmid + zdst[d];
        v = v > 0.f ? v : 0.01f * v;          // leaky_relu(0.01)
        scores[e] = v;
        lm = fmaxf(lm, v);
    }
    __shared__ float sh[256];
    sh[threadIdx.x] = lm; __syncthreads();
    for (int s = 128; s > 0; s >>= 1) {
        if (threadIdx.x < s) sh[threadIdx.x] = fmaxf(sh[threadIdx.x], sh[threadIdx.x + s]);
        __syncthreads();
    }
    if (threadIdx.x == 0) pmax[blockIdx.x] = sh[0];
}

__global__ __launch_bounds__(256) void reduce_max_kernel(
    const float* __restrict__ pmax, float* __restrict__ sc, int np)
{
    float lm = -3.4e38f;
    for (int i = threadIdx.x; i < np; i += 256) lm = fmaxf(lm, pmax[i]);
    __shared__ float sh[256];
    sh[threadIdx.x] = lm; __syncthreads();
    for (int s = 128; s > 0; s >>= 1) {
        if (threadIdx.x < s) sh[threadIdx.x] = fmaxf(sh[threadIdx.x], sh[threadIdx.x + s]);
        __syncthreads();
    }
    if (threadIdx.x == 0) sc[1] = sh[0];      // global max
}

// ============ exp(score-max) in place + per-block sums ======================
__global__ __launch_bounds__(256) void exp_sum_kernel(
    float* __restrict__ scores, const float* __restrict__ sc,
    float* __restrict__ psum, int nE)
{
    const float gmax = sc[1];
    float ls = 0.f;
    const int stride = gridDim.x * blockDim.x;
    for (int e = blockIdx.x * blockDim.x + threadIdx.x; e < nE; e += stride) {
        const float ex = __expf(scores[e] - gmax);
        scores[e] = ex;
        ls += ex;
    }
    __shared__ float sh[256];
    sh[threadIdx.x] = ls; __syncthreads();
    for (int s = 128; s > 0; s >>= 1) {
        if (threadIdx.x < s) sh[threadIdx.x] += sh[threadIdx.x + s];
        __syncthreads();
    }
    if (threadIdx.x == 0) psum[blockIdx.x] = sh[0];
}

__global__ __launch_bounds__(256) void reduce_sum_kernel(
    const float* __restrict__ psum, float* __restrict__ sc, int np)
{
    float ls = 0.f;
    for (int i = threadIdx.x; i < np; i += 256) ls += psum[i];
    __shared__ float sh[256];
    sh[threadIdx.x] = ls; __syncthreads();
    for (int s = 128; s > 0; s >>= 1) {
        if (threadIdx.x < s) sh[threadIdx.x] += sh[threadIdx.x + s];
        __syncthreads();
    }
    if (threadIdx.x == 0) sc[2] = 1.0f / sh[0];  // inverse denominator
}

// ============ final scatter: out[src] += alpha_e * x[dst] ===================
__global__ __launch_bounds__(256) void edge_out_kernel(
    const int* __restrict__ trip, const float* __restrict__ scores,
    const float* __restrict__ sc, const float* __restrict__ x,
    float* __restrict__ out, int nE)
{
    const float invden = sc[2];
    const long total = (long)nE * 16;
    const long stride = (long)gridDim.x * blockDim.x;
    for (long t = (long)blockIdx.x * blockDim.x + threadIdx.x; t < total; t += stride) {
        const int e = (int)(t >> 4);
        const int q = ((int)t & 15) * 4;
        const int s = trip[3 * e + 0];
        const int d = trip[3 * e + 2];
        const float alpha = scores[e] * invden;
        const float4 xv = *(const float4*)(x + d * DDIM + q);
        float* op = out + s * DDIM + q;
        atomicAdd(op + 0, xv.x * alpha);
        atomicAdd(op + 1, xv.y * alpha);
        atomicAdd(op + 2, xv.z * alpha);
        atomicAdd(op + 3, xv.w * alpha);
    }
}

extern "C" void kernel_launch(void* const* d_in, const int* in_sizes, int n_in,
                              void* d_out, int out_size, void* d_ws, size_t ws_size,
                              hipStream_t stream) {
    const float* emb_e  = (const float*)d_in[0];
    const float* re_att = (const float*)d_in[1];
    const float* W1     = (const float*)d_in[2];
    const float* W2     = (const float*)d_in[3];
    const float* W3     = (const float*)d_in[4];
    const float* u      = (const float*)d_in[5];
    const float* enw    = (const float*)d_in[6];
    const float* rew    = (const float*)d_in[7];
    const float* respec = (const float*)d_in[8];
    const int*   trip   = (const int*)d_in[9];

    const int nE = in_sizes[9] / 3;
    const int n  = N_ENT_C;

    // workspace layout (floats); total ~42.8 MB
    float* ws = (float*)d_ws;
    float* xA     = ws;                    // 3,200,000
    float* xB     = xA + 3200000;          // 3,200,000
    float* msg    = xB + 3200000;          // 3,200,000
    float* scores = msg + 3200000;         // 1,000,000
    float* zsrc   = scores + 1000000;      // 50,000
    float* zdst   = zsrc + 50000;          // 50,000
    float* pmax   = zdst + 50000;          // 1,024
    float* psum   = pmax + 1024;           // 1,024
    float* vsrc   = psum + 1024;           // 64
    float* vdst   = vsrc + 64;             // 64
    float* sc     = vdst + 64;             // [0]=zmid [1]=gmax [2]=1/denom

    const size_t xbytes = (size_t)n * DDIM * sizeof(float);
    const dim3 blk(256);
    const int gemmGrid = (n + 127) / 128;
    const int scatGrid = 2048;
    const int redGrid  = 1024;

    // ---- layer 1 ----
    hipMemsetAsync(msg, 0, xbytes, stream);
    edge_msg_kernel<<<scatGrid, blk, 0, stream>>>(trip, re_att, emb_e, msg, nE);
    wgc_gemm_tanh<<<gemmGrid, blk, 0, stream>>>(emb_e, msg, W1, xA, n);
    // ---- layer 2 ----
    hipMemsetAsync(msg, 0, xbytes, stream);
    edge_msg_kernel<<<scatGrid, blk, 0, stream>>>(trip, re_att, xA, msg, nE);
    wgc_gemm_tanh<<<gemmGrid, blk, 0, stream>>>(xA, msg, W2, xB, n);
    // ---- layer 3 ----
    hipMemsetAsync(msg, 0, xbytes, stream);
    edge_msg_kernel<<<scatGrid, blk, 0, stream>>>(trip, re_att, xB, msg, nE);
    wgc_gemm_tanh<<<gemmGrid, blk, 0, stream>>>(xB, msg, W3, xA, n);   // final x in xA

    // ---- attention ----
    prep_kernel<<<1, 64, 0, stream>>>(u, enw, rew, respec, vsrc, vdst, sc);
    z_kernel<<<(n + 255) / 256, blk, 0, stream>>>(xA, vsrc, vdst, zsrc, zdst, n);
    score_max_kernel<<<redGrid, blk, 0, stream>>>(trip, zsrc, zdst, sc, scores, pmax, nE);
    reduce_max_kernel<<<1, blk, 0, stream>>>(pmax, sc, redGrid);
    exp_sum_kernel<<<redGrid, blk, 0, stream>>>(scores, sc, psum, nE);
    reduce_sum_kernel<<<1, blk, 0, stream>>>(psum, sc, redGrid);

    // ---- output scatter ----
    hipMemsetAsync(d_out, 0, (size_t)out_size * sizeof(float), stream);
    edge_out_kernel<<<scatGrid, blk, 0, stream>>>(trip, scores, sc, xA, (float*)d_out, nE);
}